// TopologicalRegularizer_77902116815248
// MI455X (gfx1250) — compile-verified
//
#include <hip/hip_runtime.h>

typedef unsigned long long ull;
typedef float v2f __attribute__((ext_vector_type(2)));
typedef float v8f __attribute__((ext_vector_type(8)));

#define IMG_PIX 65536u      // 256*256
#define BATCH   8
#define TILE    2048        // elements per LDS sort tile
#define SHORT_T_C 0.08f

// ---------------------------------------------------------------------------
// 1) Build sort keys: f = 1 - prob*mask ; key = (bits(f)<<32) | local_idx.
//    f in [0,1] -> positive IEEE floats -> bit order == numeric order, and the
//    low 32 bits give the stable tie-break by index (matches np stable argsort).
// ---------------------------------------------------------------------------
__global__ void topo_build_keys(const float* __restrict__ prob,
                                const float* __restrict__ mask,
                                ull* __restrict__ keys,
                                float* __restrict__ f) {
  unsigned i = blockIdx.x * blockDim.x + threadIdx.x;   // 0 .. B*65536-1
  float fv = 1.0f - prob[i] * mask[i];
  f[i] = fv;
  keys[i] = ((ull)__float_as_uint(fv) << 32) | (ull)(i & (IMG_PIX - 1u));
}

// ---------------------------------------------------------------------------
// 2a) Bitonic: global compare-exchange step (distance j >= TILE).
// ---------------------------------------------------------------------------
__global__ void topo_bitonic_global(ull* __restrict__ keys, unsigned k, unsigned j) {
  unsigned gi = blockIdx.x * blockDim.x + threadIdx.x;
  unsigned ii = gi & (IMG_PIX - 1u);
  unsigned l  = ii ^ j;
  if (l <= ii) return;
  unsigned base = gi & ~(IMG_PIX - 1u);
  ull a = keys[base + ii];
  ull b = keys[base + l];
  bool asc = ((ii & k) == 0u);          // k==65536 -> ii&k==0 always (ascending)
  bool sw  = asc ? (a > b) : (a < b);
  if (sw) { keys[base + ii] = b; keys[base + l] = a; }
}

// ---------------------------------------------------------------------------
// 2b) Bitonic: LDS-resident stages. Handles all (k in [k_start,k_end], j<=1024)
//     inside a 2048-element tile. Tile is staged with the CDNA5 async
//     global->LDS path (ASYNCcnt) and written back with async LDS->global.
// ---------------------------------------------------------------------------
__global__ __launch_bounds__(1024)
void topo_bitonic_lds(ull* __restrict__ keys, unsigned k_start, unsigned k_end) {
  __shared__ ull tile[TILE];
  const unsigned t = threadIdx.x;                       // 0..1023
  const unsigned tile_elem_base = blockIdx.x * TILE;    // global element base
  ull* gbase = keys + tile_elem_base;

  unsigned lds0 = (unsigned)(size_t)&tile[t];
  unsigned lds1 = (unsigned)(size_t)&tile[t + 1024];
  unsigned off0 = t * 8u;
  unsigned off1 = (t + 1024u) * 8u;

  // async copy: global -> LDS (tracked by ASYNCcnt)
  asm volatile("global_load_async_to_lds_b64 %0, %1, %2"
               :: "v"(lds0), "v"(off0), "s"(gbase) : "memory");
  asm volatile("global_load_async_to_lds_b64 %0, %1, %2"
               :: "v"(lds1), "v"(off1), "s"(gbase) : "memory");
  asm volatile("s_wait_asynccnt 0x0" ::: "memory");
  __syncthreads();

  const unsigned img_local_base = tile_elem_base & (IMG_PIX - 1u);
  for (unsigned k = k_start; k <= k_end; k <<= 1) {
    unsigned jmax = (k >> 1) < 1024u ? (k >> 1) : 1024u;
    for (unsigned j = jmax; j >= 1u; j >>= 1) {
      unsigned i = ((t & ~(j - 1u)) << 1) | (t & (j - 1u));
      unsigned l = i | j;
      bool asc = (((img_local_base + i) & k) == 0u);
      ull a = tile[i];
      ull b = tile[l];
      bool sw = asc ? (a > b) : (a < b);
      if (sw) { tile[i] = b; tile[l] = a; }
      __syncthreads();
    }
  }

  // async copy back: LDS -> global
  asm volatile("s_wait_dscnt 0x0" ::: "memory");
  asm volatile("global_store_async_from_lds_b64 %0, %1, %2"
               :: "v"(off0), "v"(lds0), "s"(gbase) : "memory");
  asm volatile("global_store_async_from_lds_b64 %0, %1, %2"
               :: "v"(off1), "v"(lds1), "s"(gbase) : "memory");
  asm volatile("s_wait_asynccnt 0x0" ::: "memory");
}

// ---------------------------------------------------------------------------
// 3) rank[pixel] = position in sorted order (per image). Comparing ranks ==
//    comparing (f, idx) lexicographically (elder rule).
// ---------------------------------------------------------------------------
__global__ void topo_build_rank(const ull* __restrict__ keys,
                                unsigned* __restrict__ rank) {
  unsigned gi = blockIdx.x * blockDim.x + threadIdx.x;
  unsigned img = gi >> 16;
  unsigned pos = gi & (IMG_PIX - 1u);
  unsigned pix = (unsigned)(keys[gi] & 0xFFFFFFFFull);
  rank[(img << 16) + pix] = pos;
}

// ---------------------------------------------------------------------------
// 4) Serial union-find per image (one workgroup/image, parent[] in LDS:
//    256 KB fits CDNA5's 320 KB WGP LDS). Invariant: root == birth pixel,
//    because the younger root (larger rank) is always attached under the
//    older root. Every union emits one finite bar (birth=young root,
//    death=current pixel) — exactly the reference's elder-rule bookkeeping.
// ---------------------------------------------------------------------------
__device__ __forceinline__ int uf_find(int* parent, int x) {
  int root = x;
  int pr = parent[root];
  while (pr != root) { root = pr; pr = parent[root]; }
  while (parent[x] != root) { int nx = parent[x]; parent[x] = root; x = nx; }
  return root;
}

__global__ void topo_union_find(const ull* __restrict__ keys,
                                const float* __restrict__ f,
                                const unsigned* __restrict__ rank,
                                float* __restrict__ losses) {
  extern __shared__ int parent[];                 // 65536 ints = 256 KB LDS
  const unsigned img = blockIdx.x;
  const ull*      k  = keys + (size_t)img * IMG_PIX;
  const float*    fi = f    + (size_t)img * IMG_PIX;
  const unsigned* ri = rank + (size_t)img * IMG_PIX;

  for (unsigned i = threadIdx.x; i < IMG_PIX; i += blockDim.x) parent[i] = -1;
  __syncthreads();
  if (threadIdx.x != 0) return;

  float shortp = 0.0f;
  int nbars = 0;
  for (int t = 0; t < (int)IMG_PIX; ++t) {
    __builtin_prefetch(&k[t + 32], 0, 0);         // global_prefetch_b8
    ull key = k[t];
    int p = (int)(unsigned)(key & 0xFFFFFFFFull);
    float fp = __uint_as_float((unsigned)(key >> 32));
    parent[p] = p;                                // singleton: root == birth == p
    int r = p >> 8, c = p & 255;
    int nb0 = (r > 0)   ? p - 256 : -1;
    int nb1 = (r < 255) ? p + 256 : -1;
    int nb2 = (c > 0)   ? p - 1   : -1;
    int nb3 = (c < 255) ? p + 1   : -1;
    int nbs[4] = {nb0, nb1, nb2, nb3};            // same order as reference
    #pragma unroll
    for (int e = 0; e < 4; ++e) {
      int q = nbs[e];
      if (q < 0 || parent[q] == -1) continue;
      int rp = uf_find(parent, p);
      int rq = uf_find(parent, q);
      if (rp == rq) continue;
      int young, old;
      if (ri[rp] > ri[rq]) { young = rp; old = rq; }
      else                 { young = rq; old = rp; }
      float life = fp - fi[young];                // death value - birth value
      float d = SHORT_T_C - life;
      if (d > 0.0f) shortp += d;
      ++nbars;
      parent[young] = old;                        // keeps root==birth invariant
    }
  }
  float excess = (float)nbars - 4.0f;
  if (excess < 0.0f) excess = 0.0f;
  losses[img] = excess * excess * 0.3f + shortp * 0.7f;
}

// ---------------------------------------------------------------------------
// 5) Final reduction of the 8 per-image losses via V_WMMA_F32_16X16X4_F32.
//    B = all-ones, so D[:,0] column-sum == sum of all A entries regardless of
//    the exact A lane layout; column N=0 lives at lanes 0 and 16.
// ---------------------------------------------------------------------------
__global__ void topo_reduce_wmma(const float* __restrict__ losses,
                                 float* __restrict__ out) {
  int lane = threadIdx.x;         // 32 threads, one wave, EXEC all ones
  int m  = lane & 15;
  int hi = lane >> 4;
  int base = (m < 2) ? (m * 4 + hi * 2) : 0;      // always in-bounds
  float lx = losses[base];
  float ly = losses[base + 1];
  v2f a;
  a.x = (m < 2) ? lx : 0.0f;                      // A[0][0..3]=l0..3, A[1][0..3]=l4..7
  a.y = (m < 2) ? ly : 0.0f;
  v2f b; b.x = 1.0f; b.y = 1.0f;                  // B == all ones
  v8f c = {};
  c = __builtin_amdgcn_wmma_f32_16x16x4_f32(false, a, false, b,
                                            (short)0, c, false, false);
  float s = c[0] + c[1] + c[2] + c[3] + c[4] + c[5] + c[6] + c[7];
  float other = __shfl_xor(s, 16, 32);            // lane0 <-> lane16
  if (lane == 0) out[0] = (s + other) * (300.0f / 8.0f);
}

// ---------------------------------------------------------------------------
// Host-side orchestration (graph-capture safe: kernel launches only).
// ---------------------------------------------------------------------------
extern "C" void kernel_launch(void* const* d_in, const int* in_sizes, int n_in,
                              void* d_out, int out_size, void* d_ws, size_t ws_size,
                              hipStream_t stream) {
  const float* prob = (const float*)d_in[0];
  const float* mask = (const float*)d_in[1];
  float* out = (float*)d_out;

  char* ws = (char*)d_ws;
  ull*      keys   = (ull*)ws;                              // 4 MB
  float*    f      = (float*)(ws + 4ull * 1024 * 1024);     // 2 MB
  unsigned* rank   = (unsigned*)(ws + 6ull * 1024 * 1024);  // 2 MB
  float*    losses = (float*)(ws + 8ull * 1024 * 1024);     // 32 B

  const unsigned N = BATCH * IMG_PIX;

  topo_build_keys<<<N / 256, 256, 0, stream>>>(prob, mask, keys, f);

  // bitonic sort, per image, ascending over 64-bit keys
  topo_bitonic_lds<<<N / TILE, 1024, 0, stream>>>(keys, 2u, 2048u);
  for (unsigned k = 4096u; k <= 65536u; k <<= 1) {
    for (unsigned j = k >> 1; j >= 2048u; j >>= 1)
      topo_bitonic_global<<<N / 256, 256, 0, stream>>>(keys, k, j);
    topo_bitonic_lds<<<N / TILE, 1024, 0, stream>>>(keys, k, k);
  }

  topo_build_rank<<<N / 256, 256, 0, stream>>>(keys, rank);
  topo_union_find<<<BATCH, 256, IMG_PIX * sizeof(int), stream>>>(keys, f, rank, losses);
  topo_reduce_wmma<<<1, 32, 0, stream>>>(losses, out);
}